// ChebConv_61710090109749
// MI455X (gfx1250) — compile-verified
//
#include <hip/hip_runtime.h>

typedef __attribute__((ext_vector_type(2))) float v2f;
typedef __attribute__((ext_vector_type(8))) float v8f;

#define V_DIM 12288
#define CIN   16
#define COUT  32
#define KCHEB 5
#define WAVES_PER_BLOCK 8
#define KCHUNK (V_DIM / WAVES_PER_BLOCK)  // 1536 K-values per wave
#define KSTEPS (KCHUNK / 4)               // 384 WMMA steps per wave

// One Chebyshev matmul pass: Tout = alpha * (Tin @ L^T) - beta * Tsub
// Tin/Tsub/Tout: (CIN, V) row-major.  L: (V, V) row-major.
// Each block owns one 16-wide column tile v0..v0+15; 8 waves split K.
__global__ __launch_bounds__(256) void cheb_pass(
    const float* __restrict__ Tin, const float* __restrict__ L,
    const float* __restrict__ Tsub, float* __restrict__ Tout,
    float alpha, float beta)
{
    __shared__ float red[WAVES_PER_BLOCK * 256];

    const int v0   = blockIdx.x * 16;
    const int wave = threadIdx.x >> 5;
    const int lane = threadIdx.x & 31;
    const int m    = lane & 15;          // A row / B column within the tile
    const int kh   = (lane >> 4) * 2;    // lanes 16..31 hold K+2,K+3

    // A: T_prev[m, u + kh + {0,1}]  (contiguous float2 per lane; L2-resident)
    const float* aPtr = Tin + (size_t)m * V_DIM + wave * KCHUNK + kh;
    // B: L[v0+m, u + kh + {0,1}]^T  (contiguous float2 per lane; streamed)
    const float* bPtr = L + (size_t)(v0 + m) * V_DIM + wave * KCHUNK + kh;

    v8f c = {};
    #pragma unroll 4
    for (int s = 0; s < KSTEPS; ++s) {
        v2f a = *(const v2f*)(aPtr + s * 4);
        v2f b = __builtin_nontemporal_load((const v2f*)(bPtr + s * 4));
        // 8 args: (neg_a, A, neg_b, B, c_mod, C, reuse_a, reuse_b)
        c = __builtin_amdgcn_wmma_f32_16x16x4_f32(
            false, a, false, b, (short)0, c, false, false);
    }

    // C/D layout: lanes 0-15 -> rows 0..7 (vgpr r), lanes 16-31 -> rows 8..15
    const int rbase = (lane >> 4) * 8;
    const int n     = lane & 15;
    #pragma unroll
    for (int r = 0; r < 8; ++r)
        red[wave * 256 + (rbase + r) * 16 + n] = c[r];
    __syncthreads();

    // Thread t reduces element (row=t/16, col=t%16) across the 8 waves and
    // applies the Chebyshev recurrence in the same writeback.
    float sum = 0.f;
    #pragma unroll
    for (int w = 0; w < WAVES_PER_BLOCK; ++w)
        sum += red[w * 256 + threadIdx.x];

    const int row = threadIdx.x >> 4;
    const int col = threadIdx.x & 15;
    const size_t oidx = (size_t)row * V_DIM + v0 + col;
    Tout[oidx] = alpha * sum - beta * Tsub[oidx];
}

// out[o, v] = bias[o] + sum_k sum_i W[k,i,o] * T_k[i, v]
// T0 = x (input), T1..T4 live consecutively in workspace.
__global__ __launch_bounds__(256) void cheb_combine(
    const float* __restrict__ T0, const float* __restrict__ Tw,
    const float* __restrict__ W, const float* __restrict__ bias,
    float* __restrict__ out)
{
    __shared__ float wl[KCHEB * CIN * COUT];   // 2560 floats = 10 KB
    for (int i = threadIdx.x; i < KCHEB * CIN * COUT; i += blockDim.x)
        wl[i] = W[i];
    __syncthreads();

    const int v = blockIdx.x * blockDim.x + threadIdx.x;

    float acc[COUT];
    #pragma unroll
    for (int o = 0; o < COUT; ++o) acc[o] = bias[o];

    for (int k = 0; k < KCHEB; ++k) {
        const float* T = (k == 0) ? T0 : (Tw + (size_t)(k - 1) * CIN * V_DIM);
        #pragma unroll
        for (int i = 0; i < CIN; ++i) {
            const float t = T[(size_t)i * V_DIM + v];
            const float* wrow = &wl[(k * CIN + i) * COUT];
            #pragma unroll
            for (int o = 0; o < COUT; ++o)
                acc[o] = fmaf(t, wrow[o], acc[o]);
        }
    }
    #pragma unroll
    for (int o = 0; o < COUT; ++o)
        out[(size_t)o * V_DIM + v] = acc[o];
}

extern "C" void kernel_launch(void* const* d_in, const int* in_sizes, int n_in,
                              void* d_out, int out_size, void* d_ws, size_t ws_size,
                              hipStream_t stream)
{
    const float* x    = (const float*)d_in[0];   // (16, V)
    const float* L    = (const float*)d_in[1];   // (V, V)
    const float* W    = (const float*)d_in[2];   // (5, 16, 32)
    const float* bias = (const float*)d_in[3];   // (32,)
    float* out = (float*)d_out;                  // (32, V)

    float* T = (float*)d_ws;                     // T1..T4, each CIN*V floats
    const size_t tsz = (size_t)CIN * V_DIM;

    dim3 blk(256);
    dim3 grid(V_DIM / 16);   // 768 column tiles

    // T1 = x @ L^T
    cheb_pass<<<grid, blk, 0, stream>>>(x,           L, x,           T + 0 * tsz, 1.f, 0.f);
    // T2 = 2*(T1 @ L^T) - x
    cheb_pass<<<grid, blk, 0, stream>>>(T + 0 * tsz, L, x,           T + 1 * tsz, 2.f, 1.f);
    // T3 = 2*(T2 @ L^T) - T1
    cheb_pass<<<grid, blk, 0, stream>>>(T + 1 * tsz, L, T + 0 * tsz, T + 2 * tsz, 2.f, 1.f);
    // T4 = 2*(T3 @ L^T) - T2
    cheb_pass<<<grid, blk, 0, stream>>>(T + 2 * tsz, L, T + 1 * tsz, T + 3 * tsz, 2.f, 1.f);

    // out = einsum(kiv,kio->ov) + bias
    cheb_combine<<<dim3(V_DIM / 256), blk, 0, stream>>>(x, T, W, bias, out);
}